// AdaptiveEdgeSmoothing_55791625175595
// MI455X (gfx1250) — compile-verified
//
#include <hip/hip_runtime.h>

typedef float v2f __attribute__((ext_vector_type(2)));
typedef float v8f __attribute__((ext_vector_type(8)));

#define TILE   16
#define WAVES  8        // waves per block = horizontally adjacent 16x16 tiles
#define PATCH  20       // TILE + 2*2 halo (blur needs +/-2, laplacian +/-1)
#define PSTR   20

__global__ __launch_bounds__(256)
void AdaptiveEdgeSmoothing_kernel(const float* __restrict__ mask,
                                  const float* __restrict__ blur_strength,
                                  const float* __restrict__ edge_sensitivity,
                                  const float* __restrict__ final_threshold,
                                  float* __restrict__ out,
                                  int H, int W)
{
    __shared__ float patch[WAVES][PATCH * PSTR];   // per-wave 20x20 halo tile
    __shared__ float tbuf [WAVES][PATCH * TILE];   // per-wave horizontal 5-sums (20 rows x 16 cols)

    const int lane = threadIdx.x & 31;
    const int wave = threadIdx.x >> 5;

    const int bidx = blockIdx.z;                       // batch sample
    const int ty0  = blockIdx.y * TILE;                // tile row origin
    const int tx0  = (blockIdx.x * WAVES + wave) * TILE; // tile col origin

    // per-sample scalars (uniform per block -> scalar loads)
    const float bs = blur_strength[bidx];
    const float es = edge_sensitivity[bidx];
    const float ft = final_threshold[bidx];
    const float edge_thr = 0.5f * es;
    const float bf = bs * (1.0f / 3.0f);

    const float* img = mask + (size_t)bidx * H * W;

    // ---- load 20x20 zero-padded patch (SAME conv => zero padding) ----
    float* P = patch[wave];
    for (int i = lane; i < PATCH * PATCH; i += 32) {
        int pr = i / PATCH, pc = i - pr * PATCH;
        int gy = ty0 + pr - 2;
        int gx = tx0 + pc - 2;
        float v = 0.0f;
        if (gy >= 0 && gy < H && gx >= 0 && gx < W)
            v = img[(size_t)gy * W + gx];
        P[pr * PSTR + pc] = v;
    }
    __syncthreads();

    // ---- horizontal 5-sums: T[r][c] = sum P[r][c..c+4], r in [0,20), c in [0,16) ----
    float* T = tbuf[wave];
    for (int i = lane; i < PATCH * TILE; i += 32) {
        int r = i >> 4, c = i & 15;
        const float* row = &P[r * PSTR + c];
        T[i] = row[0] + row[1] + row[2] + row[3] + row[4];
    }
    __syncthreads();

    // WMMA C/D layout: VGPR r, lanes 0-15 -> row r, lanes 16-31 -> row r+8; col = lane&15
    const int n     = lane & 15;
    const int rbase = (lane >> 4) * 8;

    // ---- Laplacian ingredients: 3-wide row sums for 10 rows, plus centers ----
    float rs3[10];
    #pragma unroll
    for (int j = 0; j < 10; ++j) {
        const float* row = &P[(rbase + 1 + j) * PSTR + (n + 1)];
        rs3[j] = row[0] + row[1] + row[2];
    }
    float cent[8];
    #pragma unroll
    for (int r = 0; r < 8; ++r)
        cent[r] = P[(rbase + 2 + r) * PSTR + (n + 2)];

    // ---- vertical 5-sum via WMMA: D(16x16) = Cv(16x20) x T(20x16), Cv banded 0/1 ----
    v8f acc = {0.f, 0.f, 0.f, 0.f, 0.f, 0.f, 0.f, 0.f};
#if defined(__has_builtin) && __has_builtin(__builtin_amdgcn_wmma_f32_16x16x4_f32)
    {
        // A 16x4 f32 layout: lanes 0-15 hold M=0..15 {K=k0,k0+1}; lanes 16-31 {K=k0+2,k0+3}
        const int mA   = lane & 15;
        const int ksub = (lane >> 4) * 2;
        #pragma unroll
        for (int kc = 0; kc < 5; ++kc) {
            const int k0 = kc * 4 + ksub;
            v2f a, bmat;
            a.x = (k0     >= mA && k0     <= mA + 4) ? 1.0f : 0.0f;
            a.y = (k0 + 1 >= mA && k0 + 1 <= mA + 4) ? 1.0f : 0.0f;
            // B 4x16 f32: V0 lanes0-15=row k0, lanes16-31=row k0+1; V1 rows k0+2,k0+3
            const int brow = kc * 4 + (lane >> 4);
            bmat.x = T[brow * TILE + n];
            bmat.y = T[(brow + 2) * TILE + n];
            acc = __builtin_amdgcn_wmma_f32_16x16x4_f32(
                /*neg_a=*/false, a, /*neg_b=*/false, bmat,
                /*c_mod=*/(short)0, acc, /*reuse_a=*/false, /*reuse_b=*/false);
        }
    }
#else
    {
        #pragma unroll
        for (int r = 0; r < 8; ++r) {
            const int m = rbase + r;
            acc[r] = T[m * TILE + n] + T[(m + 1) * TILE + n] + T[(m + 2) * TILE + n]
                   + T[(m + 3) * TILE + n] + T[(m + 4) * TILE + n];
        }
    }
#endif

    // ---- fuse: edge mask, blend, select, threshold; write 8 rows per lane ----
    float* obase = out + (size_t)bidx * H * W;
    #pragma unroll
    for (int r = 0; r < 8; ++r) {
        const int m  = rbase + r;
        const float c    = cent[r];
        const float sum3 = rs3[r] + rs3[r + 1] + rs3[r + 2];
        const float lap  = 9.0f * c - sum3;                 // 8*c - (8 neighbors)
        const float em   = (fabsf(lap) > edge_thr) ? 1.0f : 0.0f;
        const float sb   = acc[r] * (1.0f / 25.0f);         // 5x5 box blur
        const float sm   = c * (1.0f - bf) + sb * bf;
        const float res  = c * (1.0f - em) + sm * em;
        const float fm   = (res > ft) ? 1.0f : 0.0f;
        obase[(size_t)(ty0 + m) * W + (tx0 + n)] = fm;
    }
}

extern "C" void kernel_launch(void* const* d_in, const int* in_sizes, int n_in,
                              void* d_out, int out_size, void* d_ws, size_t ws_size,
                              hipStream_t stream) {
    const float* mask = (const float*)d_in[0];
    const float* bs   = (const float*)d_in[1];
    const float* es   = (const float*)d_in[2];
    const float* ft   = (const float*)d_in[3];
    float* out = (float*)d_out;

    const int H = 1024, W = 1024;
    const int B = in_sizes[0] / (H * W);   // 16

    dim3 grid(W / (TILE * WAVES), H / TILE, B);   // (8, 64, 16)
    dim3 block(256);
    AdaptiveEdgeSmoothing_kernel<<<grid, block, 0, stream>>>(mask, bs, es, ft, out, H, W);
}